// Capsule_32521492365403
// MI455X (gfx1250) — compile-verified
//
#include <hip/hip_runtime.h>
#include <math.h>

typedef float v2f __attribute__((ext_vector_type(2)));
typedef float v8f __attribute__((ext_vector_type(8)));

#define NC   32
#define DC   64
#define NIN  1024
#define DIN  512
#define BATCH 64

// ---------------------------------------------------------------------------
// GEMM: uhat[b][nc][i][dc] = sum_k U[b][i][k] * W[k][nc*64+dc]
// Block = 128 (i) x 64 (one capsule's dc columns), 256 threads = 8 waves (4x2).
// f32 WMMA 16x16x4, K accumulated in LDS chunks of 64.
// ---------------------------------------------------------------------------
__global__ __launch_bounds__(256) void capsule_gemm(const float* __restrict__ U,
                                                    const float* __restrict__ W,
                                                    float* __restrict__ uhat) {
    __shared__ float ldsA[128 * 68];   // padded stride: 68*4B = 272B (16B aligned, bank step 4)
    __shared__ float ldsB[64 * 64];

    const int t    = threadIdx.x;
    const int wave = t >> 5;
    const int lane = t & 31;
    const int half = lane >> 4;        // 0/1: lane half
    const int l16  = lane & 15;
    const int wm   = wave >> 1;        // 0..3 -> M sub-tile (32 rows)
    const int wn   = wave & 1;         // 0..1 -> N sub-tile (32 cols)

    const int nc = blockIdx.x;         // capsule = 64-column tile
    const int rt = blockIdx.y;         // row tile (128 rows of N_in)
    const int b  = blockIdx.z;

    const float* Ab = U + ((size_t)b * NIN + (size_t)rt * 128) * DIN;
    const float* Bb = W + (size_t)nc * DC;

    v8f acc[2][2] = {};

    const int lr = t >> 4;             // 0..15 (staging row)
    const int lc = (t & 15) << 2;      // 0..60 step 4 (staging col, float4)

    for (int k0 = 0; k0 < DIN; k0 += 64) {
        // Stage A chunk: 128 x 64
        #pragma unroll
        for (int p = 0; p < 8; ++p) {
            int row = p * 16 + lr;
            float4 v = *(const float4*)(Ab + (size_t)row * DIN + k0 + lc);
            *(float4*)(&ldsA[row * 68 + lc]) = v;
        }
        // Stage B chunk: 64 x 64
        #pragma unroll
        for (int p = 0; p < 4; ++p) {
            int kr = p * 16 + lr;
            float4 v = *(const float4*)(Bb + (size_t)(k0 + kr) * (NC * DC) + lc);
            *(float4*)(&ldsB[kr * 64 + lc]) = v;
        }
        __syncthreads();

        const int aR0 = (wm * 32 + l16) * 68;       // A frag: M = tile_m + l16
        const int aR1 = (wm * 32 + 16 + l16) * 68;
        const int bC0 = wn * 32 + l16;              // B frag: N = tile_n + l16

        #pragma unroll
        for (int kk = 0; kk < 64; kk += 4) {
            const int ka = kk + 2 * half;           // K split over lane halves
            v2f a0 = *(const v2f*)(&ldsA[aR0 + ka]);
            v2f a1 = *(const v2f*)(&ldsA[aR1 + ka]);
            v2f b0, b1;
            b0.x = ldsB[ka * 64 + bC0];
            b0.y = ldsB[(ka + 1) * 64 + bC0];
            b1.x = ldsB[ka * 64 + bC0 + 16];
            b1.y = ldsB[(ka + 1) * 64 + bC0 + 16];
            acc[0][0] = __builtin_amdgcn_wmma_f32_16x16x4_f32(false, a0, false, b0, (short)0, acc[0][0], false, false);
            acc[0][1] = __builtin_amdgcn_wmma_f32_16x16x4_f32(false, a0, false, b1, (short)0, acc[0][1], false, false);
            acc[1][0] = __builtin_amdgcn_wmma_f32_16x16x4_f32(false, a1, false, b0, (short)0, acc[1][0], false, false);
            acc[1][1] = __builtin_amdgcn_wmma_f32_16x16x4_f32(false, a1, false, b1, (short)0, acc[1][1], false, false);
        }
        __syncthreads();
    }

    // Epilogue: scatter into transposed layout uhat[b][nc][row][dc].
    // D layout (ISA 7.12.2): vgpr r, lane L -> M = r + 8*(L/16), N = L%16.
    float* outBase = uhat + ((size_t)b * NC + nc) * NIN * DC;
    const int rowT = rt * 128 + wm * 32;
    #pragma unroll
    for (int mi = 0; mi < 2; ++mi) {
        #pragma unroll
        for (int r = 0; r < 8; ++r) {
            int row = rowT + mi * 16 + r + 8 * half;
            float* dst = outBase + (size_t)row * DC + wn * 32 + l16;
            dst[0]  = acc[mi][0][r];
            dst[16] = acc[mi][1][r];
        }
    }
}

// ---------------------------------------------------------------------------
// Weighted sum + squash: dst[b,n,:] = squash( sum_i c[b,n,i] * uhat[b,n,i,:] )
// One block per (b,n); fully coalesced streaming of the 1024x64 tile.
// ---------------------------------------------------------------------------
__global__ __launch_bounds__(256) void capsule_wsum(const float* __restrict__ uhat,
                                                    const float* __restrict__ c,
                                                    float* __restrict__ dst,
                                                    int uniform) {
    __shared__ float red[256];
    __shared__ float rs;
    const int t  = threadIdx.x;
    const int bn = blockIdx.x;          // b*32 + n
    const int d  = t & 63;
    const int g  = t >> 6;              // 4 i-groups
    const float* up = uhat + (size_t)bn * NIN * DC;
    const float* cp = c + (size_t)bn * NIN;

    float acc = 0.f;
    if (uniform) {
        for (int i = g; i < NIN; i += 4)
            acc += up[(size_t)i * DC + d];
        acc *= (1.0f / 32.0f);          // softmax of zeros over 32 capsules
    } else {
        for (int i = g; i < NIN; i += 4)
            acc += cp[i] * up[(size_t)i * DC + d];
    }
    red[t] = acc;
    __syncthreads();
    if (t < 64) red[t] = red[t] + red[t + 64] + red[t + 128] + red[t + 192];
    __syncthreads();
    if (t == 0) {
        float ss = 0.f;
        #pragma unroll
        for (int j = 0; j < 64; ++j) ss += red[j] * red[j];
        rs = rsqrtf(ss + 1e-7f);        // repo's squash: x / sqrt(||x||^2 + eps)
    }
    __syncthreads();
    if (t < 64) dst[(size_t)bn * DC + t] = red[t] * rs;
}

// ---------------------------------------------------------------------------
// Agreement: bt[b][i][n] = dot(outputs[b,n,:], uhat[b,n,i,:])  (b is REPLACED)
// One block per (b,n); float4 loads, 16-lane shuffle reduction.
// ---------------------------------------------------------------------------
__global__ __launch_bounds__(256) void capsule_agree(const float* __restrict__ uhat,
                                                     const float* __restrict__ outv,
                                                     float* __restrict__ bt) {
    __shared__ float o[64];
    const int t  = threadIdx.x;
    const int bn = blockIdx.x;
    const int b  = bn >> 5;
    const int n  = bn & 31;
    if (t < 64) o[t] = outv[(size_t)bn * DC + t];
    __syncthreads();
    const float* up = uhat + (size_t)bn * NIN * DC;
    const int il = t >> 4;              // 16 rows per step
    const int q  = t & 15;              // 16 float4 segments per row
    for (int i0 = 0; i0 < NIN; i0 += 16) {
        int i = i0 + il;
        float4 u = *(const float4*)(up + (size_t)i * DC + q * 4);
        float p = u.x * o[q * 4] + u.y * o[q * 4 + 1] + u.z * o[q * 4 + 2] + u.w * o[q * 4 + 3];
        #pragma unroll
        for (int off = 8; off >= 1; off >>= 1)
            p += __shfl_xor(p, off, 16);
        if (q == 0)
            bt[((size_t)b * NIN + i) * NC + n] = p;
    }
}

// ---------------------------------------------------------------------------
// Softmax over the capsule axis: c[b,n,i] = softmax_n(bt[b,i,n])
// One thread per (b,i); contiguous 32-float read, scattered 8MB write (cheap).
// ---------------------------------------------------------------------------
__global__ __launch_bounds__(256) void capsule_softmax(const float* __restrict__ bt,
                                                       float* __restrict__ c) {
    const int tid = blockIdx.x * 256 + threadIdx.x;   // 64*1024 threads
    const int b = tid >> 10;
    const int i = tid & 1023;
    const float* p = bt + (size_t)tid * NC;
    float v[NC];
    float m = -1e30f;
    #pragma unroll
    for (int n = 0; n < NC; ++n) { v[n] = p[n]; m = fmaxf(m, v[n]); }
    float s = 0.f;
    #pragma unroll
    for (int n = 0; n < NC; ++n) { v[n] = __expf(v[n] - m); s += v[n]; }
    const float inv = 1.0f / s;
    #pragma unroll
    for (int n = 0; n < NC; ++n)
        c[((size_t)b * NC + n) * NIN + i] = v[n] * inv;
}

// ---------------------------------------------------------------------------
extern "C" void kernel_launch(void* const* d_in, const int* in_sizes, int n_in,
                              void* d_out, int out_size, void* d_ws, size_t ws_size,
                              hipStream_t stream) {
    const float* U = (const float*)d_in[0];   // [64, 1024, 512] fp32
    const float* W = (const float*)d_in[1];   // [1, 512, 2048]  fp32
    float* out = (float*)d_out;               // [64, 32, 64]    fp32

    char* ws = (char*)d_ws;
    float* uhat = (float*)ws;                                    // 512 MB: [B,NC,NIN,DC]
    float* outv = (float*)(ws + 536870912ull);                   // 512 KB: [B,NC,DC]
    float* bt   = (float*)(ws + 536870912ull + 1048576ull);      // 8 MB:   [B,NIN,NC]
    float* cbuf = (float*)(ws + 536870912ull + 1048576ull + 8388608ull); // 8 MB: [B,NC,NIN]

    dim3 gGemm(NC, NIN / 128, BATCH);
    capsule_gemm<<<gGemm, 256, 0, stream>>>(U, W, uhat);

    // Routing iteration 0: c = softmax(0) = uniform
    capsule_wsum<<<BATCH * NC, 256, 0, stream>>>(uhat, cbuf, outv, 1);
    capsule_agree<<<BATCH * NC, 256, 0, stream>>>(uhat, outv, bt);
    capsule_softmax<<<(BATCH * NIN) / 256, 256, 0, stream>>>(bt, cbuf);

    // Routing iteration 1
    capsule_wsum<<<BATCH * NC, 256, 0, stream>>>(uhat, cbuf, outv, 0);
    capsule_agree<<<BATCH * NC, 256, 0, stream>>>(uhat, outv, bt);
    capsule_softmax<<<(BATCH * NIN) / 256, 256, 0, stream>>>(bt, cbuf);

    // Routing iteration 2 (final): write squashed outputs straight to d_out
    capsule_wsum<<<BATCH * NC, 256, 0, stream>>>(uhat, cbuf, out, 0);
}